// deformable_LKA_11003706212444
// MI455X (gfx1250) — compile-verified
//
#include <hip/hip_runtime.h>

typedef __attribute__((ext_vector_type(2))) float v2f;
typedef __attribute__((ext_vector_type(8))) float v8f;

#define B_  4
#define C_  64
#define H_  64
#define W_  64
#define HW  4096   // H_*W_

// ---------------------------------------------------------------------------
// Prep: transpose conv weights into [K][Npad] (zero-padded cols) for coalesced
// WMMA B-fragment loads.
//   Bt0 : [576][64]   from w_off0 [50][64][3][3]  (pad 50->64)
//   Bt1 : [576][112]  from w_off1 [98][64][3][3]  (pad 98->112)
//   BtPW: [64][64]    from w_pw   [64][64]
// ---------------------------------------------------------------------------
__global__ void prep_weights(const float* __restrict__ w_off0,
                             const float* __restrict__ w_off1,
                             const float* __restrict__ w_pw,
                             float* __restrict__ Bt0,
                             float* __restrict__ Bt1,
                             float* __restrict__ BtPW)
{
    const int total0 = 576 * 64;
    const int total1 = 576 * 112;
    const int totalp = 64 * 64;
    const int total  = total0 + total1 + totalp;
    for (int i = blockIdx.x * blockDim.x + threadIdx.x; i < total;
         i += gridDim.x * blockDim.x) {
        if (i < total0) {
            int k = i / 64, n = i % 64;
            Bt0[k * 64 + n] = (n < 50) ? w_off0[n * 576 + k] : 0.0f;
        } else if (i < total0 + total1) {
            int j = i - total0;
            int k = j / 112, n = j % 112;
            Bt1[k * 112 + n] = (n < 98) ? w_off1[n * 576 + k] : 0.0f;
        } else {
            int j = i - total0 - total1;
            int k = j / 64, n = j % 64;
            BtPW[k * 64 + n] = w_pw[n * 64 + k];
        }
    }
}

// ---------------------------------------------------------------------------
// Dense 3x3 conv (pad=1) via V_WMMA_F32_16X16X4_F32.
//   grid = (HW/16, B), block = (32, Npad/16)  -- one wave per N-tile.
//   LDS im2col chunk: 16 channels x 9 taps x 16 positions = 144x16 f32 (9KB),
//   stored [k][m] so A-fragment reads are bank-conflict free.
// A 16x4 layout: lane&15 = M; lanes<16 hold K=0,1; lanes>=16 hold K=2,3.
// C/D layout: lane&15 = N; VGPR v holds M=v (lo half) / M=v+8 (hi half).
// ---------------------------------------------------------------------------
__global__ void conv3x3_wmma(const float* __restrict__ xin,  // [B][64][H][W]
                             const float* __restrict__ Bt,   // [576][Npad]
                             const float* __restrict__ bias, // [Cout]
                             float* __restrict__ out,        // [B][Cout][H][W]
                             int Cout, int Npad)
{
    __shared__ float Atile[144 * 16];

    const int b    = blockIdx.y;
    const int p0   = blockIdx.x * 16;          // 16 consecutive x in one row
    const int y    = p0 >> 6;
    const int x0   = p0 & 63;
    const int lane = threadIdx.x;
    const int tid  = threadIdx.y * 32 + lane;
    const int nthr = blockDim.y * 32;
    const int n0   = threadIdx.y * 16;
    const int mn   = lane & 15;
    const int hi   = lane >> 4;

    v8f acc = {};

    for (int c0 = 0; c0 < C_; c0 += 16) {
        __syncthreads();
        // im2col fill: k_local = cl*9 + (ky*3+kx), 2304 elements
        for (int i = tid; i < 144 * 16; i += nthr) {
            int kl = i >> 4;
            int m  = i & 15;
            int cl = kl / 9;
            int t  = kl - cl * 9;
            int ky = t / 3, kx = t - ky * 3;
            int yy = y - 1 + ky;
            int xx = x0 + m - 1 + kx;
            float v = 0.0f;
            if (yy >= 0 && yy < H_ && xx >= 0 && xx < W_)
                v = xin[((b * C_ + c0 + cl) * H_ + yy) * W_ + xx];
            Atile[i] = v;
        }
        __syncthreads();

        const float* Bp = Bt + (c0 * 9) * Npad + n0 + mn;
        for (int ks = 0; ks < 144; ks += 4) {
            v2f a, bf;
            a.x  = Atile[(ks + hi * 2 + 0) * 16 + mn];
            a.y  = Atile[(ks + hi * 2 + 1) * 16 + mn];
            bf.x = Bp[(ks + hi * 2 + 0) * Npad];
            bf.y = Bp[(ks + hi * 2 + 1) * Npad];
            acc = __builtin_amdgcn_wmma_f32_16x16x4_f32(
                false, a, false, bf, (short)0, acc, false, false);
        }
    }

    const int n = n0 + mn;
    if (n < Cout) {
        const float bb = bias[n];
        const size_t base = ((size_t)(b * Cout + n)) * HW + p0 + hi * 8;
#pragma unroll
        for (int v = 0; v < 8; ++v)
            out[base + v] = acc[v] + bb;
    }
}

// ---------------------------------------------------------------------------
// Deformable depthwise conv. One block = 256 threads = one (b,y) row half
// (32 x-positions). Phase 1: tap geometry (shared across channels) -> LDS.
// Phase 2: 64 channels x 32 x, lanes sweep x for coalesced gathers.
// ---------------------------------------------------------------------------
template <int K, int DIL>
__global__ void deform_dw(const float* __restrict__ xin,  // [B][64][H][W]
                          const float* __restrict__ off,  // [B][2*K*K][H][W]
                          const float* __restrict__ wdw,  // [64][K*K]
                          const float* __restrict__ bdw,  // [64]
                          float* __restrict__ out)        // [B][64][H][W]
{
    constexpr int K2  = K * K;
    constexpr int PAD = (K / 2) * DIL;
    __shared__ int   s_pk[K2 * 32];
    __shared__ float s_w0[K2 * 32];
    __shared__ float s_w1[K2 * 32];
    __shared__ float s_w2[K2 * 32];
    __shared__ float s_w3[K2 * 32];

    const int chunk = blockIdx.x;          // [0, B*H*2)
    const int xc    = (chunk & 1) * 32;
    const int row   = chunk >> 1;          // b*H + y
    const int y     = row & 63;
    const int b     = row >> 6;
    const int tid   = threadIdx.x;

    for (int e = tid; e < K2 * 32; e += 256) {
        int k2 = e >> 5;
        int xl = e & 31;
        int x  = xc + xl;
        int ky = k2 / K, kx = k2 - ky * K;
        float oy = off[(((b * 2 * K2) + 2 * k2 + 0) * H_ + y) * W_ + x];
        float ox = off[(((b * 2 * K2) + 2 * k2 + 1) * H_ + y) * W_ + x];
        float py = (float)(y - PAD + DIL * ky) + oy;
        float px = (float)(x - PAD + DIL * kx) + ox;
        float fy = floorf(py), fx = floorf(px);
        float wy1 = py - fy, wx1 = px - fx;
        float wy0 = 1.0f - wy1, wx0 = 1.0f - wx1;
        float vy0 = (fy >= 0.0f && fy <= (float)(H_ - 1)) ? 1.0f : 0.0f;
        float vy1 = (fy + 1.0f >= 0.0f && fy + 1.0f <= (float)(H_ - 1)) ? 1.0f : 0.0f;
        float vx0 = (fx >= 0.0f && fx <= (float)(W_ - 1)) ? 1.0f : 0.0f;
        float vx1 = (fx + 1.0f >= 0.0f && fx + 1.0f <= (float)(W_ - 1)) ? 1.0f : 0.0f;
        int ya = min(max((int)fy, 0), H_ - 1);
        int yb = min(max((int)fy + 1, 0), H_ - 1);
        int xa = min(max((int)fx, 0), W_ - 1);
        int xb = min(max((int)fx + 1, 0), W_ - 1);
        s_pk[e] = ya | (yb << 8) | (xa << 16) | (xb << 24);
        s_w0[e] = wy0 * wx0 * vy0 * vx0;
        s_w1[e] = wy0 * wx1 * vy0 * vx1;
        s_w2[e] = wy1 * wx0 * vy1 * vx0;
        s_w3[e] = wy1 * wx1 * vy1 * vx1;
    }
    __syncthreads();

    for (int it = 0; it < 8; ++it) {
        int idx = it * 256 + tid;         // 64 channels x 32 x
        int c   = idx >> 5;
        int xl  = idx & 31;
        const float* plane = xin + ((size_t)(b * C_ + c)) * HW;
        const float* wk    = wdw + c * K2;
        float acc = 0.0f;
        for (int k2 = 0; k2 < K2; ++k2) {
            int e  = k2 * 32 + xl;
            int pk = s_pk[e];
            int ya = pk & 255, yb = (pk >> 8) & 255;
            int xa = (pk >> 16) & 255, xb = (pk >> 24) & 255;
            float p00 = plane[ya * W_ + xa];
            float p01 = plane[ya * W_ + xb];
            float p10 = plane[yb * W_ + xa];
            float p11 = plane[yb * W_ + xb];
            float sv = s_w0[e] * p00 + s_w1[e] * p01 + s_w2[e] * p10 + s_w3[e] * p11;
            acc = fmaf(wk[k2], sv, acc);
        }
        out[((size_t)(b * C_ + c)) * HW + y * W_ + xc + xl] = acc + bdw[c];
    }
}

// ---------------------------------------------------------------------------
// 1x1 conv (M=positions, N=K=64) via WMMA, fused bias + u*attn.
// ---------------------------------------------------------------------------
__global__ void pointwise_wmma(const float* __restrict__ attn, // [B][64][HW]
                               const float* __restrict__ BtPW, // [64][64]
                               const float* __restrict__ bpw,  // [64]
                               const float* __restrict__ u,    // [B][64][HW]
                               float* __restrict__ out)
{
    const int b    = blockIdx.y;
    const int p0   = blockIdx.x * 16;
    const int lane = threadIdx.x;
    const int n0   = threadIdx.y * 16;
    const int mn   = lane & 15;
    const int hi   = lane >> 4;

    v8f acc = {};
    const float* Ap = attn + (size_t)b * C_ * HW + p0 + mn;
    const float* Bp = BtPW + n0 + mn;
    for (int ks = 0; ks < 64; ks += 4) {
        v2f a, bf;
        a.x  = Ap[(size_t)(ks + hi * 2 + 0) * HW];
        a.y  = Ap[(size_t)(ks + hi * 2 + 1) * HW];
        bf.x = Bp[(ks + hi * 2 + 0) * 64];
        bf.y = Bp[(ks + hi * 2 + 1) * 64];
        acc = __builtin_amdgcn_wmma_f32_16x16x4_f32(
            false, a, false, bf, (short)0, acc, false, false);
    }
    const int o = n0 + mn;
    const float bb = bpw[o];
#pragma unroll
    for (int v = 0; v < 8; ++v) {
        int m = v + hi * 8;
        size_t idx = ((size_t)(b * C_ + o)) * HW + p0 + m;
        out[idx] = (acc[v] + bb) * u[idx];
    }
}

// ---------------------------------------------------------------------------
extern "C" void kernel_launch(void* const* d_in, const int* in_sizes, int n_in,
                              void* d_out, int out_size, void* d_ws, size_t ws_size,
                              hipStream_t stream)
{
    (void)in_sizes; (void)n_in; (void)out_size; (void)ws_size;

    const float* x      = (const float*)d_in[0];
    const float* w_off0 = (const float*)d_in[1];
    const float* b_off0 = (const float*)d_in[2];
    const float* w_dw0  = (const float*)d_in[3];
    const float* b_dw0  = (const float*)d_in[4];
    const float* w_off1 = (const float*)d_in[5];
    const float* b_off1 = (const float*)d_in[6];
    const float* w_dw1  = (const float*)d_in[7];
    const float* b_dw1  = (const float*)d_in[8];
    const float* w_pw   = (const float*)d_in[9];
    const float* b_pw   = (const float*)d_in[10];
    float* out = (float*)d_out;

    // workspace layout (floats), total ~18.5 MB
    float* ws    = (float*)d_ws;
    float* off0  = ws;                       // 4*50*4096
    float* attn0 = off0  + (size_t)B_ * 50 * HW;   // 4*64*4096
    float* off1  = attn0 + (size_t)B_ * C_ * HW;   // 4*98*4096
    float* attn1 = off1  + (size_t)B_ * 98 * HW;   // 4*64*4096
    float* Bt0   = attn1 + (size_t)B_ * C_ * HW;   // 576*64
    float* Bt1   = Bt0   + 576 * 64;               // 576*112
    float* BtPW  = Bt1   + 576 * 112;              // 64*64

    prep_weights<<<dim3(128), dim3(256), 0, stream>>>(w_off0, w_off1, w_pw,
                                                      Bt0, Bt1, BtPW);

    // off0 = conv3x3(x), Cout=50 (pad 64)
    conv3x3_wmma<<<dim3(HW / 16, B_), dim3(32, 4), 0, stream>>>(
        x, Bt0, b_off0, off0, 50, 64);

    // attn0 = deform_dw 5x5 dil=1
    deform_dw<5, 1><<<dim3(B_ * H_ * 2), dim3(256), 0, stream>>>(
        x, off0, w_dw0, b_dw0, attn0);

    // off1 = conv3x3(attn0), Cout=98 (pad 112)
    conv3x3_wmma<<<dim3(HW / 16, B_), dim3(32, 7), 0, stream>>>(
        attn0, Bt1, b_off1, off1, 98, 112);

    // attn1 = deform_dw 7x7 dil=3
    deform_dw<7, 3><<<dim3(B_ * H_ * 2), dim3(256), 0, stream>>>(
        attn0, off1, w_dw1, b_dw1, attn1);

    // out = x * (pointwise(attn1) + b_pw)
    pointwise_wmma<<<dim3(HW / 16, B_), dim3(32, 4), 0, stream>>>(
        attn1, BtPW, b_pw, x, out);
}